// BinaryMasking_17145509445656
// MI455X (gfx1250) — compile-verified
//
#include <hip/hip_runtime.h>
#include <hip/hip_bf16.h>
#include <stdint.h>

// ---------------- problem constants (compile-time, from reference) ----------
#define TDIM    16
#define HW      4096          // H*W
#define NTOK    65536         // N = T*H*W
#define BROW    128           // B
#define EPSC    0.01f
#define CLAMP_LO 1e-6f
#define SRC_ALPHA 3.0f
// EVENT_ALPHA = 2.0  ->  * 0.5f below

#define THREADS 1024
#define CHUNK   4096                    // == HW, so event term is constant per chunk
#define NCHUNK  (NTOK / CHUNK)          // 16
#define EPT     4                       // elements/thread/chunk (one b128 per thread)

// ---------------- gfx1250 async global->LDS staging -------------------------
#if defined(__HIP_DEVICE_COMPILE__)
#if __has_builtin(__builtin_amdgcn_global_load_async_to_lds_b128)
#define HAVE_ASYNC_LDS 1
#endif
#if __has_builtin(__builtin_amdgcn_s_wait_asynccnt)
#define HAVE_WAIT_ASYNC 1
#endif
#endif

#ifdef HAVE_ASYNC_LDS
typedef int v4i __attribute__((ext_vector_type(4)));
typedef __attribute__((address_space(1))) v4i g_v4i;   // global (AS1) int4
typedef __attribute__((address_space(3))) v4i l_v4i;   // LDS    (AS3) int4
#endif

__device__ __forceinline__ void stage_b128(const float* g, float* l) {
#ifdef HAVE_ASYNC_LDS
  // global_load_async_to_lds_b128: per-lane 16B global -> per-lane LDS slot
  __builtin_amdgcn_global_load_async_to_lds_b128(
      (g_v4i*)(uintptr_t)g,   // strip const, reinterpret as AS1 int4*
      (l_v4i*)l,              // generic->AS3 addrspacecast
      0, 0);
#else
  *(float4*)l = *(const float4*)g;   // fallback: plain copy through VGPRs
#endif
}

template <int N>
__device__ __forceinline__ void wait_async() {
#ifdef HAVE_WAIT_ASYNC
  __builtin_amdgcn_s_wait_asynccnt(N);
#elif defined(HAVE_ASYNC_LDS)
  if constexpr (N == 0) asm volatile("s_wait_asynccnt 0x0" ::: "memory");
  else                  asm volatile("s_wait_asynccnt 0x1" ::: "memory");
#endif
}

// ---------------- helpers ----------------------------------------------------
__device__ __forceinline__ float clampu(float u) {
  return fminf(fmaxf(u, CLAMP_LO), 1.0f - CLAMP_LO);
}
// monotone float -> uint key (ascending order preserved)
__device__ __forceinline__ unsigned fkey(float f) {
  unsigned b = __float_as_uint(f);
  return (b & 0x80000000u) ? ~b : (b | 0x80000000u);
}
// per-row rate parameters (identical math in both kernels => identical bits)
__device__ __forceinline__ void rate_params(const float* Ur, int m, int b,
                                            int* K, float* dR) {
  float ur = clampu(Ur[0]);
  float L  = EPSC + (float)b * ((1.0f - 2.0f * EPSC) / (float)(BROW - 1));
  float U  = L + ur; if (U >= 1.0f) U -= 1.0f;          // (L + u) % 1.0
  float R  = (m == 0) ? expf(logf(U) / SRC_ALPHA) : (1.0f - U);
  *K  = (int)(R * (float)NTOK);                          // trunc like .long()
  *dR = expf(logf(U) * (1.0f / SRC_ALPHA - 1.0f)) / SRC_ALPHA;
}

// =============================================================================
// Kernel 1: exact per-row K-th-largest threshold via 2-level 16-bit radix
// select. One workgroup per (matrix m, row b). LDS: 65536-bin histogram
// (256 KB) + 32 KB double-buffered async staging.
// =============================================================================
__global__ void __launch_bounds__(THREADS)
topk_select_kernel(const float* __restrict__ Uw, const float* __restrict__ Ue,
                   const float* __restrict__ Ur,
                   unsigned* __restrict__ thrOut, int* __restrict__ cntOut) {
  extern __shared__ char dyn[];
  unsigned* hist  = (unsigned*)dyn;                      // 65536 bins
  float*    stage = (float*)(dyn + 65536u * 4u);         // 2 * CHUNK floats

  __shared__ unsigned scanbuf[THREADS];
  __shared__ float    le[NCHUNK];
  __shared__ int      sK, sSkip;
  __shared__ unsigned sThrHi, sCntHi, sThrLo, sCntLo;

  const int tid = threadIdx.x;
  const int blk = blockIdx.x;
  const int m   = blk / BROW;
  const int b   = blk % BROW;

  if (tid == 0) {
    int K; float dR;
    rate_params(Ur, m, b, &K, &dR);
    sK = K;
    sSkip = (K <= 0);
    if (K <= 0) { thrOut[blk] = 0xFFFFFFFFu; cntOut[blk] = 0; }
  }
  if (tid < NCHUNK) {
    float ue = clampu(Ue[b * TDIM + tid]);
    le[tid] = (m == 0) ? (0.5f * logf(ue)) : (0.5f * log1pf(-ue));
  }
  __syncthreads();
  if (sSkip) return;
  const int K = sK;
  const float* row = Uw + ((size_t)m * BROW + b) * (size_t)NTOK;

  for (int pass = 0; pass < 2; ++pass) {
    for (int i = tid; i < 65536; i += THREADS) hist[i] = 0u;
    __syncthreads();
    const unsigned thrHi = sThrHi;                 // valid in pass 1 only

    // ---- stream the row through LDS (async, double buffered) ----
    stage_b128(row + (size_t)tid * EPT, stage + tid * EPT);
    for (int c = 0; c < NCHUNK; ++c) {
      const int cur = c & 1;
      if (c + 1 < NCHUNK) {
        stage_b128(row + (size_t)(c + 1) * CHUNK + (size_t)tid * EPT,
                   stage + ((c + 1) & 1) * CHUNK + tid * EPT);
        wait_async<1>();
      } else {
        wait_async<0>();
      }
      const float lev = le[c];
#pragma unroll
      for (int j = 0; j < EPT; ++j) {
        float u    = clampu(stage[cur * CHUNK + tid * EPT + j]);
        unsigned k = fkey(logf(u) + lev);
        if (pass == 0)                 atomicAdd(&hist[k >> 16], 1u);
        else if ((k >> 16) == thrHi)   atomicAdd(&hist[k & 0xFFFFu], 1u);
      }
    }
    __syncthreads();

    // ---- descending cumulative walk to locate the K-th bin ----
    const unsigned targetK = (pass == 0) ? (unsigned)K : ((unsigned)K - sCntHi);
    const int base = 65535 - tid * 64;             // thread owns 64 bins, high->low
    unsigned partial = 0;
    for (int i = 0; i < 64; ++i) partial += hist[base - i];
    scanbuf[tid] = partial;
    __syncthreads();
    for (int off = 1; off < THREADS; off <<= 1) {  // inclusive Hillis-Steele
      unsigned v = scanbuf[tid];
      if (tid >= off) v += scanbuf[tid - off];
      __syncthreads();
      scanbuf[tid] = v;
      __syncthreads();
    }
    unsigned c0 = scanbuf[tid] - partial;          // count strictly above my range
    for (int i = 0; i < 64; ++i) {
      unsigned h = hist[base - i];
      if (c0 < targetK && c0 + h >= targetK) {     // exactly one (tid,i) fires
        if (pass == 0) { sThrHi = (unsigned)(base - i); sCntHi = c0; }
        else           { sThrLo = (unsigned)(base - i); sCntLo = c0; }
      }
      c0 += h;
    }
    __syncthreads();
  }

  if (tid == 0) {
    thrOut[blk] = (sThrHi << 16) | sThrLo;         // exact K-th largest key
    cntOut[blk] = (int)(sCntHi + sCntLo);          // # strictly greater
  }
}

// =============================================================================
// Kernel 2: emit masks with exact stable (index-ordered) tie-breaking, plus
// the dR plane. One workgroup per (m, b); chunks processed in index order
// with a workgroup prefix scan over tie flags.
// =============================================================================
__global__ void __launch_bounds__(THREADS)
emit_kernel(const float* __restrict__ Uw, const float* __restrict__ Ue,
            const float* __restrict__ Ur, const unsigned* __restrict__ thrIn,
            const int* __restrict__ cntIn, float* __restrict__ out) {
  __shared__ float    stage[2 * CHUNK];            // 32 KB
  __shared__ unsigned scanbuf[THREADS];
  __shared__ float    le[NCHUNK];
  __shared__ unsigned sThr;
  __shared__ int      sCntGt, sK, sEqBase;
  __shared__ float    sdR;

  const int tid = threadIdx.x;
  const int blk = blockIdx.x;
  const int m   = blk / BROW;
  const int b   = blk % BROW;

  if (tid == 0) {
    int K; float dR;
    rate_params(Ur, m, b, &K, &dR);
    sK = K; sdR = dR;
    sThr = thrIn[blk];
    sCntGt = cntIn[blk];
    sEqBase = 0;
  }
  if (tid < NCHUNK) {
    float ue = clampu(Ue[b * TDIM + tid]);
    le[tid] = (m == 0) ? (0.5f * logf(ue)) : (0.5f * log1pf(-ue));
  }
  __syncthreads();
  const unsigned thr = sThr;
  const int K = sK, cntGt = sCntGt;
  const float dR = sdR;
  const float* row  = Uw  + ((size_t)m * BROW + b) * (size_t)NTOK;
  float*       orow = out + ((size_t)m * BROW + b) * (size_t)NTOK;
  float*       drow = out + (size_t)2 * BROW * NTOK + (size_t)b * NTOK;

  stage_b128(row + (size_t)tid * EPT, stage + tid * EPT);
  for (int c = 0; c < NCHUNK; ++c) {
    const int cur = c & 1;
    if (c + 1 < NCHUNK) {
      stage_b128(row + (size_t)(c + 1) * CHUNK + (size_t)tid * EPT,
                 stage + ((c + 1) & 1) * CHUNK + tid * EPT);
      wait_async<1>();
    } else {
      wait_async<0>();
    }
    const float lev = le[c];
    bool gt[EPT], eq[EPT];
    unsigned cnt = 0;
#pragma unroll
    for (int j = 0; j < EPT; ++j) {
      float u    = clampu(stage[cur * CHUNK + tid * EPT + j]);
      unsigned k = fkey(logf(u) + lev);
      gt[j] = (k > thr);
      eq[j] = (k == thr);
      cnt  += eq[j] ? 1u : 0u;
    }
    // index-ordered exclusive prefix of tie flags across the workgroup
    scanbuf[tid] = cnt;
    __syncthreads();
    for (int off = 1; off < THREADS; off <<= 1) {
      unsigned v = scanbuf[tid];
      if (tid >= off) v += scanbuf[tid - off];
      __syncthreads();
      scanbuf[tid] = v;
      __syncthreads();
    }
    const unsigned incl  = scanbuf[tid];
    const unsigned total = scanbuf[THREADS - 1];
    const int base = sEqBase;                      // read before tid-0 update
    int run = base + (int)(incl - cnt);
    float ov[EPT];
#pragma unroll
    for (int j = 0; j < EPT; ++j) {
      bool sel = gt[j] || (eq[j] && (cntGt + run) < K);  // stable tie rank
      if (eq[j]) ++run;
      ov[j] = sel ? 1.0f : 0.0f;
    }
    const size_t off4 = (size_t)c * CHUNK + (size_t)tid * EPT;
    *(float4*)(orow + off4) = make_float4(ov[0], ov[1], ov[2], ov[3]);
    if (m == 0) *(float4*)(drow + off4) = make_float4(dR, dR, dR, dR);
    __syncthreads();
    if (tid == 0) sEqBase = base + (int)total;
    __syncthreads();
  }
}

// =============================================================================
extern "C" void kernel_launch(void* const* d_in, const int* in_sizes, int n_in,
                              void* d_out, int out_size, void* d_ws, size_t ws_size,
                              hipStream_t stream) {
  (void)in_sizes; (void)n_in; (void)out_size; (void)ws_size;
  const float* Uw = (const float*)d_in[0];   // (2, B, N)
  const float* Ue = (const float*)d_in[1];   // (B, T)
  const float* Ur = (const float*)d_in[2];   // (1,)
  float* out = (float*)d_out;                // src | tgt | dR, each B*N floats

  unsigned* thrW = (unsigned*)d_ws;
  int*      cntW = (int*)((char*)d_ws + 2 * BROW * sizeof(unsigned));

  const size_t shmem = 65536u * sizeof(unsigned) + 2u * CHUNK * sizeof(float);
  topk_select_kernel<<<dim3(2 * BROW), dim3(THREADS), shmem, stream>>>(
      Uw, Ue, Ur, thrW, cntW);
  emit_kernel<<<dim3(2 * BROW), dim3(THREADS), 0, stream>>>(
      Uw, Ue, Ur, thrW, cntW, out);
}